// MelBandMerge_67224828117616
// MI455X (gfx1250) — compile-verified
//
#include <hip/hip_runtime.h>
#include <hip/hip_bf16.h>

// MelBandMerge on gfx1250:
//   12 small GEMMs  (M = o_i in {170,172}, N = B*T = 32768, K = G = 32)
//   D = W_i (o x G)  *  x_slice (G x T)  + bias  ->  out[B,2,1025,T].
// Memory-bound: ~4.3 GFLOP vs ~320 MB traffic (269 MB output stream) ->
// ~14 us floor at 23.3 TB/s. Compute via V_WMMA_F32_16X16X4_F32
// (8 chained WMMAs per 16x16 tile, K=32).
//
// Per wave: one band, one 16-wide t-tile. x fragments (B matrix) loaded once
// (non-temporal single-use stream), reused across all output tiles.
// The o-dimension is tiled PER OUTPUT CHANNEL (two halves of w rows each,
// 6 tiles/half) so every full tile stores through one base pointer with
// immediate offsets: no per-store channel select, no 64-bit address math.
// Fully unrolled so W/bias/store addressing folds into immediates.

typedef __attribute__((ext_vector_type(2))) float v2f;
typedef __attribute__((ext_vector_type(8))) float v8f;

#define GDIM   32
#define NBAND  12
#define TLEN   4096
#define BATCH  8
#define NFREQ  1025
#define GSTRIDE (NBAND * TLEN)   // stride between g-rows of x: 49152 floats

__constant__ int c_start[NBAND] = {0,85,170,256,341,427,512,597,683,768,854,939};
__constant__ int c_width[NBAND] = {85,85,86,85,86,85,85,86,85,86,85,86};

struct BandParams {
    const float* W[NBAND];
    const float* bia[NBAND];
};

__global__ __launch_bounds__(256) void melband_wmma_kernel(
    const float* __restrict__ x,   // [B, G*NBAND, 1, T]  == [B, G, NBAND, T]
    float* __restrict__ out,       // [B, 2, NFREQ, T]
    BandParams p)
{
    const int lane      = threadIdx.x & 31;
    const int waveInBlk = threadIdx.x >> 5;
    const int waveId    = blockIdx.x * 8 + waveInBlk;   // 12*2048 waves total

    const int band  = waveId >> 11;          // /2048
    const int tTile = waveId & 2047;
    const int idx   = tTile << 4;            // base of 16 consecutive (b*T+t)
    const int bIdx  = idx >> 12;             // /4096  (tile never straddles b)
    const int t0    = idx & (TLEN - 1);

    const int hi = lane >> 4;                // half-wave select
    const int lr = lane & 15;

    const int w  = c_width[band];            // 85 or 86
    const int s0 = c_start[band];

    // ---- B matrix fragments: x slice, K=32 (8 steps of K=4), N = 16 t's ----
    // lane holds column t = t0 + lr; VGPR v of step s holds K = 4s + 2*hi + v.
    const float* xp = x + ((size_t)(bIdx * GDIM) * NBAND + band) * TLEN + t0 + lr;
    v2f bf[8];
#pragma unroll
    for (int s = 0; s < 8; ++s) {
        const int g0 = 4 * s + 2 * hi;
        bf[s].x = __builtin_nontemporal_load(xp + (size_t)g0       * GSTRIDE);
        bf[s].y = __builtin_nontemporal_load(xp + (size_t)(g0 + 1) * GSTRIDE);
    }

    const float* Wb = p.W[band];
    const float* Bb = p.bia[band];

    // per-(band, t-tile, lane) output base of channel 0; channel 1 is a
    // band-independent constant jump of NFREQ*TLEN floats.
    float* outB = out + ((size_t)bIdx * 2 * NFREQ + s0) * TLEN + t0 + lr;

#pragma unroll
    for (int cc = 0; cc < 2; ++cc) {
        const int rbase = cc * w;                         // row base in W/bias
        // per-lane base pointers; everything below folds into immediates
        const float* WpB  = Wb + (size_t)(rbase + lr) * GDIM + 2 * hi;
        const float* BbB  = Bb + rbase + 8 * hi;
        float*       dstB = outB + ((size_t)cc * NFREQ + 8 * hi) * TLEN;

        // ---- 5 full tiles: rows 0..79 of this half, always valid ----
#pragma unroll
        for (int wT = 0; wT < 5; ++wT) {
            const int wi0 = wT * 16;
            v8f acc;
#pragma unroll
            for (int r = 0; r < 8; ++r)
                acc[r] = BbB[wi0 + r];                    // bias -> C accum
#pragma unroll
            for (int s = 0; s < 8; ++s) {
                v2f a;
                a.x = WpB[wi0 * GDIM + 4 * s];
                a.y = WpB[wi0 * GDIM + 4 * s + 1];
                acc = __builtin_amdgcn_wmma_f32_16x16x4_f32(
                    false, a, false, bf[s], (short)0, acc, false, false);
            }
#pragma unroll
            for (int r = 0; r < 8; ++r)
                __builtin_nontemporal_store(acc[r],
                    dstB + (size_t)(wi0 + r) * TLEN);
        }

        // ---- partial tile: rows 80..95, valid while wi < w (5 or 6 rows) ----
        {
            const int wi0 = 80;
            const int wiL = wi0 + lr;                     // 80..95
            // clamp for memory safety only: D row m depends solely on A row m,
            // so garbage rows are simply never stored.
            const float* Wp = Wb + (size_t)(rbase + (wiL < w ? wiL : w - 1)) * GDIM
                              + 2 * hi;
            v8f acc;
#pragma unroll
            for (int r = 0; r < 8; ++r) {
                const int wi = wi0 + 8 * hi + r;
                acc[r] = Bb[rbase + (wi < w ? wi : w - 1)];
            }
#pragma unroll
            for (int s = 0; s < 8; ++s) {
                v2f a;
                a.x = Wp[4 * s];
                a.y = Wp[4 * s + 1];
                acc = __builtin_amdgcn_wmma_f32_16x16x4_f32(
                    false, a, false, bf[s], (short)0, acc, false, false);
            }
#pragma unroll
            for (int r = 0; r < 8; ++r) {
                const int wi = wi0 + 8 * hi + r;
                if (wi < w)
                    __builtin_nontemporal_store(acc[r],
                        dstB + (size_t)(wi0 + r) * TLEN);
            }
        }
    }
}

extern "C" void kernel_launch(void* const* d_in, const int* in_sizes, int n_in,
                              void* d_out, int out_size, void* d_ws, size_t ws_size,
                              hipStream_t stream) {
    (void)in_sizes; (void)n_in; (void)out_size; (void)d_ws; (void)ws_size;

    BandParams p;
    for (int i = 0; i < NBAND; ++i) {
        p.W[i]   = (const float*)d_in[1 + 2 * i];
        p.bia[i] = (const float*)d_in[2 + 2 * i];
    }
    const float* x = (const float*)d_in[0];
    float* out     = (float*)d_out;

    const int totalWaves = NBAND * ((BATCH * TLEN) / 16);  // 24576
    const int blocks     = totalWaves / 8;                 // 3072 blocks x 256 thr
    melband_wmma_kernel<<<blocks, 256, 0, stream>>>(x, out, p);
}